// Model_40518721470518
// MI455X (gfx1250) — compile-verified
//
#include <hip/hip_runtime.h>
#include <math.h>

// ---------------------------------------------------------------------------
// Chain of 23 separable 2D resizes on (32,256,H,W) f32, 64x64 -> 47x30.
// Bandwidth-bound => keep f32; each separable 1D resize is a banded GEMM
// against a dense zero-padded interpolation matrix, executed with
// V_WMMA_F32_16X16X4_F32. K-loops are band-limited per 16-row/col tile
// (interp matrices are banded), cutting WMMA steps and L2 traffic 4-10x.
// ---------------------------------------------------------------------------

#define NPLANES 8192              // 32*256
#define CAP_PER_PLANE 18576       // max H*W over the chain = 108*172

typedef __attribute__((ext_vector_type(2))) float v2f;
typedef __attribute__((ext_vector_type(8))) float v8f;

#if defined(__has_builtin)
#if __has_builtin(__builtin_amdgcn_wmma_f32_16x16x4_f32)
#define HAVE_WMMA_F32 1
#endif
#endif
#ifndef HAVE_WMMA_F32
#define HAVE_WMMA_F32 0
#endif

static __device__ __forceinline__ int imin(int a, int b) { return a < b ? a : b; }
static __device__ __forceinline__ int imax(int a, int b) { return a > b ? a : b; }

#define CUBIC_A (-0.75f)
static __device__ __forceinline__ float cubic1(float t) {
    return ((CUBIC_A + 2.f) * t - (CUBIC_A + 3.f)) * t * t + 1.f;
}
static __device__ __forceinline__ float cubic2(float t) {
    return ((CUBIC_A * t - 5.f * CUBIC_A) * t + 8.f * CUBIC_A) * t - 4.f * CUBIC_A;
}

// src coordinate for output index i (same f32 math as the weight builders)
static __device__ __forceinline__ float src_of(int i, int align, float scale) {
    return align ? (float)i * scale : ((float)i + 0.5f) * scale - 0.5f;
}

// --------------------------- nearest: fused 2D gather ----------------------
__global__ void nearest2d_kernel(const float* __restrict__ X, float* __restrict__ Y,
                                 int inH, int inW, int outH, int outW,
                                 float rH, float rW, long long total) {
    long long t = (long long)blockIdx.x * blockDim.x + threadIdx.x;
    if (t >= total) return;
    int w = (int)(t % outW);
    long long q = t / outW;
    int h = (int)(q % outH);
    long long p = q / outH;
    int ih = imin((int)floorf((float)h * rH), inH - 1);
    int iw = imin((int)floorf((float)w * rW), inW - 1);
    Y[t] = X[p * (long long)inH * inW + (long long)ih * inW + iw];
}

// ------------------- dense interpolation-matrix builders -------------------
// H weights: Wm[padOutH][padInH], row-major; row h' holds taps for output row h'.
__global__ void build_h_weights(float* __restrict__ Wm, int inH, int outH,
                                int padInH, int padOutH, int mode, int align,
                                float scale) {
    int h = blockIdx.x * blockDim.x + threadIdx.x;
    if (h >= padOutH) return;
    float* row = Wm + (size_t)h * padInH;
    for (int j = 0; j < padInH; ++j) row[j] = 0.f;
    if (h >= outH) return;
    float src = src_of(h, align, scale);
    if (mode == 1) {  // bilinear
        if (!align) src = fmaxf(src, 0.f);
        float f = floorf(src);
        float t = src - f;
        int i0 = (int)f;
        int ia = imin(imax(i0, 0), inH - 1);
        int ib = imin(imax(i0 + 1, 0), inH - 1);
        row[ia] += 1.f - t;
        row[ib] += t;
    } else {  // bicubic, 4 taps, border-clamped (clamped taps accumulate)
        float f = floorf(src);
        float t = src - f;
        int i0 = (int)f;
        float ws[4] = {cubic2(t + 1.f), cubic1(t), cubic1(1.f - t), cubic2(2.f - t)};
        for (int k = 0; k < 4; ++k) {
            int idx = imin(imax(i0 - 1 + k, 0), inH - 1);
            row[idx] += ws[k];
        }
    }
}

// W weights, transposed: Wm[padInW][padOutW]; column w' holds taps for output col w'.
__global__ void build_w_weights(float* __restrict__ Wm, int inW, int outW,
                                int padInW, int padOutW, int mode, int align,
                                float scale) {
    int w = blockIdx.x * blockDim.x + threadIdx.x;
    if (w >= padOutW) return;
    for (int k = 0; k < padInW; ++k) Wm[(size_t)k * padOutW + w] = 0.f;
    if (w >= outW) return;
    float src = src_of(w, align, scale);
    if (mode == 1) {
        if (!align) src = fmaxf(src, 0.f);
        float f = floorf(src);
        float t = src - f;
        int i0 = (int)f;
        int ia = imin(imax(i0, 0), inW - 1);
        int ib = imin(imax(i0 + 1, 0), inW - 1);
        Wm[(size_t)ia * padOutW + w] += 1.f - t;
        Wm[(size_t)ib * padOutW + w] += t;
    } else {
        float f = floorf(src);
        float t = src - f;
        int i0 = (int)f;
        float ws[4] = {cubic2(t + 1.f), cubic1(t), cubic1(1.f - t), cubic2(2.f - t)};
        for (int k = 0; k < 4; ++k) {
            int idx = imin(imax(i0 - 1 + k, 0), inW - 1);
            Wm[(size_t)idx * padOutW + w] += ws[k];
        }
    }
}

// Wave-uniform band of non-zero weight columns for output tile [lo0, lo0+15].
// Linear taps span [floor(s), floor(s)+1]; cubic [floor(s)-1, floor(s)+2];
// border-clamped taps accumulate at 0 / in-1, which stay inside the clamp.
static __device__ __forceinline__ void tap_band(int tile0, int outN, int inN,
                                                int mode, int align, float scale,
                                                int& klo, int& khi) {
    int last = imin(tile0 + 15, outN - 1);
    float s0 = src_of(tile0, align, scale);
    float s1 = src_of(last, align, scale);
    int ext = (mode == 2) ? 1 : 0;
    klo = imax((int)floorf(s0) - ext, 0);
    khi = imin((int)floorf(s1) + 1 + ext, inN - 1);
    klo &= ~3;  // WMMA K-step alignment; extra columns hold zero weights
}

// --------------------- H-pass: Y[outH][W] = A[outH][inH] * X[inH][W] -------
// One wave per 16x16 output tile. A fragment (16x4 f32, 2 VGPR): lanes 0-15
// hold K0,K1, lanes 16-31 hold K2,K3 of row M=lane&15. B fragment mirrors.
__global__ void resize_h_wmma(const float* __restrict__ X, float* __restrict__ Y,
                              const float* __restrict__ Wm,
                              int inH, int Wd, int outH, int padInH,
                              int mode, int align, float scale) {
    int lane = threadIdx.x;
    int tw = blockIdx.z * blockDim.y + threadIdx.y;
    int w0 = tw * 16;
    if (w0 >= Wd) return;  // wave-uniform, EXEC stays all-ones for live waves
    size_t plane = blockIdx.x;
    const float* x = X + plane * (size_t)inH * Wd;
    float* y = Y + plane * (size_t)outH * Wd;
    int h0 = blockIdx.y * 16;
    int m = lane & 15;
    int half = lane >> 4;
    int kb = half * 2;
    int wc = imin(w0 + m, Wd - 1);  // clamped load column (stores are guarded)
    const float* wr = Wm + (size_t)(h0 + m) * padInH;
    int klo, khi;
    tap_band(h0, outH, inH, mode, align, scale, klo, khi);
    v8f acc = {};
    for (int k0 = klo; k0 <= khi; k0 += 4) {
#if HAVE_WMMA_F32
        v2f a;
        a[0] = wr[k0 + kb];
        a[1] = wr[k0 + kb + 1];
        int kr0 = imin(k0 + kb, inH - 1);      // out-of-band weights are 0
        int kr1 = imin(k0 + kb + 1, inH - 1);
        v2f b;
        b[0] = x[(size_t)kr0 * Wd + wc];
        b[1] = x[(size_t)kr1 * Wd + wc];
        acc = __builtin_amdgcn_wmma_f32_16x16x4_f32(false, a, false, b,
                                                    (short)0, acc, false, false);
#else
        for (int kk = 0; kk < 4; ++kk) {
            int k = k0 + kk;
            float xv = x[(size_t)imin(k, inH - 1) * Wd + wc];
#pragma unroll
            for (int r = 0; r < 8; ++r)
                acc[r] += Wm[(size_t)(h0 + half * 8 + r) * padInH + k] * xv;
        }
#endif
    }
    int hb = h0 + half * 8;
    int ws = w0 + m;
#pragma unroll
    for (int r = 0; r < 8; ++r) {
        int hr = hb + r;
        if (hr < outH && ws < Wd) y[(size_t)hr * Wd + ws] = acc[r];
    }
}

// --------------- W-pass: Y[H][outW] = X[H][inW] * Wt[inW][outW] ------------
__global__ void resize_w_wmma(const float* __restrict__ X, float* __restrict__ Y,
                              const float* __restrict__ Wm,
                              int Hd, int inW, int outW, int padInW, int padOutW,
                              int mode, int align, float scale) {
    int lane = threadIdx.x;
    int tw = blockIdx.z * blockDim.y + threadIdx.y;
    int w0 = tw * 16;
    if (w0 >= outW) return;
    size_t plane = blockIdx.x;
    const float* x = X + plane * (size_t)Hd * inW;
    float* y = Y + plane * (size_t)Hd * outW;
    int h0 = blockIdx.y * 16;
    int m = lane & 15;
    int half = lane >> 4;
    int kb = half * 2;
    const float* xr = x + (size_t)imin(h0 + m, Hd - 1) * inW;  // clamped row
    int klo, khi;
    tap_band(w0, outW, inW, mode, align, scale, klo, khi);
    v8f acc = {};
    for (int k0 = klo; k0 <= khi; k0 += 4) {
#if HAVE_WMMA_F32
        v2f a;
        a[0] = xr[imin(k0 + kb, inW - 1)];      // out-of-band weight rows are 0
        a[1] = xr[imin(k0 + kb + 1, inW - 1)];
        v2f b;
        b[0] = Wm[(size_t)(k0 + kb) * padOutW + w0 + m];
        b[1] = Wm[(size_t)(k0 + kb + 1) * padOutW + w0 + m];
        acc = __builtin_amdgcn_wmma_f32_16x16x4_f32(false, a, false, b,
                                                    (short)0, acc, false, false);
#else
        for (int kk = 0; kk < 4; ++kk) {
            int k = k0 + kk;
            float wv = Wm[(size_t)k * padOutW + w0 + m];
#pragma unroll
            for (int r = 0; r < 8; ++r) {
                const float* row = x + (size_t)imin(h0 + half * 8 + r, Hd - 1) * inW;
                acc[r] += row[imin(k, inW - 1)] * wv;
            }
        }
#endif
    }
    int hb = h0 + half * 8;
    int ws = w0 + m;
#pragma unroll
    for (int r = 0; r < 8; ++r) {
        int hr = hb + r;
        if (hr < Hd && ws < outW) y[(size_t)hr * outW + ws] = acc[r];
    }
}

// ---------------------------------------------------------------------------
extern "C" void kernel_launch(void* const* d_in, const int* in_sizes, int n_in,
                              void* d_out, int out_size, void* d_ws, size_t ws_size,
                              hipStream_t stream) {
    (void)in_sizes; (void)n_in; (void)out_size; (void)ws_size;
    const float* x0 = (const float*)d_in[0];

    // workspace: two ping-pong tensors (cap each) + two weight tables
    float* bufA = (float*)d_ws;
    const size_t cap = (size_t)NPLANES * CAP_PER_PLANE;
    float* bufB = bufA + cap;
    float* WH = bufB + cap;       // up to 112*108 floats
    float* WW = WH + 65536;       // up to 172*176 floats

    struct Op { int mode, align, oH, oW; };  // mode 0=nearest,1=bilinear,2=bicubic
    static const Op ops[23] = {
        {0, 0, 16, 16},  {0, 0, 32, 32},  {0, 0, 20, 20},  {0, 0, 80, 80},
        {0, 0, 16, 24},  {0, 0, 32, 72},
        {1, 0, 16, 16},  {1, 0, 32, 32},  {1, 0, 20, 20},  {1, 0, 80, 80},
        {1, 1, 16, 24},  {1, 1, 32, 72},
        {2, 0, 16, 16},  {2, 0, 32, 32},  {2, 0, 20, 20},  {2, 0, 80, 80},
        {2, 1, 16, 24},  {2, 1, 32, 72},
        {0, 0, 54, 144}, {1, 0, 108, 172}, {1, 1, 54, 68}, {2, 0, 43, 61},
        {2, 1, 47, 30},
    };

    int H = 64, W = 64;
    float* cur = bufA;
    float* alt = bufB;
    for (int i = 0; i < 23; ++i) {
        const Op o = ops[i];
        const float* s = (i == 0) ? x0 : cur;
        float* fin = (i == 22) ? (float*)d_out : nullptr;
        if (o.mode == 0) {
            float* dst = fin ? fin : alt;
            long long total = (long long)NPLANES * o.oH * o.oW;
            float rH = (float)((double)H / (double)o.oH);
            float rW = (float)((double)W / (double)o.oW);
            int blocks = (int)((total + 255) / 256);
            nearest2d_kernel<<<blocks, 256, 0, stream>>>(s, dst, H, W, o.oH, o.oW,
                                                         rH, rW, total);
            if (!fin) { float* t = cur; cur = alt; alt = t; }
        } else {
            int padInH = (H + 3) & ~3, padOutH = (o.oH + 15) & ~15;
            int padInW = (W + 3) & ~3, padOutW = (o.oW + 15) & ~15;
            float sH = o.align
                           ? (o.oH > 1 ? (float)((double)(H - 1) / (double)(o.oH - 1)) : 0.f)
                           : (float)((double)H / (double)o.oH);
            float sW = o.align
                           ? (o.oW > 1 ? (float)((double)(W - 1) / (double)(o.oW - 1)) : 0.f)
                           : (float)((double)W / (double)o.oW);
            build_h_weights<<<(padOutH + 63) / 64, 64, 0, stream>>>(
                WH, H, o.oH, padInH, padOutH, o.mode, o.align, sH);
            build_w_weights<<<(padOutW + 63) / 64, 64, 0, stream>>>(
                WW, W, o.oW, padInW, padOutW, o.mode, o.align, sW);
            float* tmp = alt;  // H-pass intermediate (oH x W)
            {
                dim3 blk(32, 4);
                dim3 grd(NPLANES, (o.oH + 15) / 16, ((W + 15) / 16 + 3) / 4);
                resize_h_wmma<<<grd, blk, 0, stream>>>(s, tmp, WH, H, W, o.oH,
                                                       padInH, o.mode, o.align, sH);
            }
            float* dst = fin ? fin : cur;  // overwrite source buffer (dead data)
            {
                dim3 blk(32, 4);
                dim3 grd(NPLANES, (o.oH + 15) / 16, ((o.oW + 15) / 16 + 3) / 4);
                resize_w_wmma<<<grd, blk, 0, stream>>>(tmp, dst, WW, o.oH, W, o.oW,
                                                       padInW, padOutW,
                                                       o.mode, o.align, sW);
            }
            // result ends up in cur (or d_out on the last op)
        }
        H = o.oH;
        W = o.oW;
    }
}